// UHGInfoNCELoss_21328807592550
// MI455X (gfx1250) — compile-verified
//
#include <hip/hip_runtime.h>
#include <hip/hip_bf16.h>
#include <math.h>

typedef __attribute__((ext_vector_type(2))) float v2f;
typedef __attribute__((ext_vector_type(8))) float v8f;

#define UHG_EPS      1e-9f
#define TEMP_INV_NEG (-1.0f / 0.07f)
#define TILE_M       128
#define TILE_N       128
#define LDS_STRIDE   68   // 64 + 4 pad to stagger LDS banks

// ---------------------------------------------------------------------------
// Kernel 1: per-row Minkowski norms  inner = sum(x[0:D-1]^2) - x[D-1]^2
// ---------------------------------------------------------------------------
__global__ __launch_bounds__(256)
void uhg_norms_kernel(const float* __restrict__ q, const float* __restrict__ k,
                      float* __restrict__ qq, float* __restrict__ kk,
                      int N, int D) {
    int i = blockIdx.x * blockDim.x + threadIdx.x;
    if (i >= 2 * N) return;
    const float* src = (i < N) ? q : k;
    int row = (i < N) ? i : i - N;
    const float4* r4 = (const float4*)(src + (size_t)row * D);
    float s = 0.f, last = 0.f;
    int nv = D >> 2;
    for (int c = 0; c < nv; ++c) {
        float4 v = r4[c];
        s += v.x * v.x + v.y * v.y + v.z * v.z + v.w * v.w;
        if (c == nv - 1) last = v.w;
    }
    float val = s - 2.f * last * last;   // spatial^2 - time^2
    if (i < N) qq[row] = val; else kk[row] = val;
}

// online logsumexp merge: (m1,s1) <- merge((m1,s1),(m2,s2))
__device__ __forceinline__ void lse_merge(float& m1, float& s1, float m2, float s2) {
    float nm = fmaxf(m1, m2);
    s1 = s1 * __expf(m1 - nm) + s2 * __expf(m2 - nm);
    m1 = nm;
}

// ---------------------------------------------------------------------------
// Kernel 2: 128x128-tile WMMA f32 GEMM  inner[n,m] = q[n,:] . k'[m,:]
// (k' = k with last coord negated), fused arccosh epilogue + fused per-row
// logsumexp partials (one (max,sumexp) pair per row per column-block).
// Block = 256 threads = 8 waves; wave (wm,wn), wm in 0..3 owns 32 rows,
// wn in 0..1 owns 64 cols.  Accumulators: 2 row-tiles x 4 col-tiles of 16x16.
// ---------------------------------------------------------------------------
__global__ __launch_bounds__(256)
void uhg_gemm_kernel(const float* __restrict__ q, const float* __restrict__ k,
                     const float* __restrict__ qq, const float* __restrict__ kk,
                     float* __restrict__ sims, float* __restrict__ dists,
                     float* __restrict__ wsm, float* __restrict__ wss,
                     int N, int D) {
    __shared__ float lds_a[TILE_M * LDS_STRIDE];
    __shared__ float lds_b[TILE_N * LDS_STRIDE];
    __shared__ float pm[TILE_M];
    __shared__ float ps[TILE_M];

    const int m_base = blockIdx.y * TILE_M;   // query rows
    const int n_base = blockIdx.x * TILE_N;   // key cols
    const int t = threadIdx.x;

    // --- stage A (q) and B (k, sign-flipped last coord) tiles into LDS ---
    const int nq4 = D >> 2;                        // float4s per row (16)
    const int iters = (TILE_M * nq4) / 256;        // 8
    for (int it = 0; it < iters; ++it) {
        int idx = t + it * 256;
        int row = idx / nq4;
        int c4  = idx % nq4;
        float4 va = *(const float4*)(q + (size_t)(m_base + row) * D + 4 * c4);
        *(float4*)&lds_a[row * LDS_STRIDE + 4 * c4] = va;
        float4 vb = *(const float4*)(k + (size_t)(n_base + row) * D + 4 * c4);
        if (4 * c4 + 3 == D - 1) vb.w = -vb.w;     // timelike sign flip
        *(float4*)&lds_b[row * LDS_STRIDE + 4 * c4] = vb;
    }
    __syncthreads();

    const int wave = t >> 5;
    const int lane = t & 31;
    const int wm   = wave >> 1;   // 0..3 : row group (32 rows)
    const int wn   = wave & 1;    // 0..1 : col group (64 cols)
    const int lm   = lane & 15;   // row (A) / col (B) within 16
    const int lh   = lane >> 4;   // K-pair / row-half selector

    v8f acc[2][4];
    const v8f vzero = {0.f, 0.f, 0.f, 0.f, 0.f, 0.f, 0.f, 0.f};
#pragma unroll
    for (int rt = 0; rt < 2; ++rt)
#pragma unroll
        for (int ct = 0; ct < 4; ++ct) acc[rt][ct] = vzero;

    for (int k0 = 0; k0 < D; k0 += 4) {
        v2f av[2], bv[4];
#pragma unroll
        for (int rt = 0; rt < 2; ++rt)
            av[rt] = *(const v2f*)&lds_a[(32 * wm + 16 * rt + lm) * LDS_STRIDE + k0 + 2 * lh];
#pragma unroll
        for (int ct = 0; ct < 4; ++ct)
            bv[ct] = *(const v2f*)&lds_b[(64 * wn + 16 * ct + lm) * LDS_STRIDE + k0 + 2 * lh];
#pragma unroll
        for (int rt = 0; rt < 2; ++rt)
#pragma unroll
            for (int ct = 0; ct < 4; ++ct)
                acc[rt][ct] = __builtin_amdgcn_wmma_f32_16x16x4_f32(
                    false, av[rt], false, bv[ct], (short)0, acc[rt][ct], false, false);
    }

    // --- epilogue: arccosh distance + similarity (NT stores) + row-lse ---
    float kkv[4];
#pragma unroll
    for (int ct = 0; ct < 4; ++ct)
        kkv[ct] = kk[n_base + 64 * wn + 16 * ct + lm];

    float rm[2][8], rs[2][8];
#pragma unroll
    for (int rt = 0; rt < 2; ++rt) {
#pragma unroll
        for (int e = 0; e < 8; ++e) {
            int rl  = 32 * wm + 16 * rt + e + 8 * lh;   // row in tile
            int row = m_base + rl;
            float qqv = qq[row];
            size_t roff = (size_t)row * N + n_base + 64 * wn + lm;

            float sim[4];
#pragma unroll
            for (int ct = 0; ct < 4; ++ct) {
                float inner = acc[rt][ct][e];
                // 1/denom = rsqrt(max(|qq*kk|, eps))  (single v_rsq_f32)
                float inv   = __builtin_amdgcn_rsqf(fmaxf(fabsf(qqv * kkv[ct]), UHG_EPS));
                float r     = fmaxf(fabsf(inner) * inv, 1.0f + UHG_EPS);
                // acosh(r) = log(r + sqrt((r-1)*(r+1)))   (XLA-style, f32)
                float d     = __logf(r + __builtin_amdgcn_sqrtf((r - 1.f) * (r + 1.f)));
                float sm    = d * TEMP_INV_NEG;
                sim[ct]     = sm;
                __builtin_nontemporal_store(sm, sims  + roff + 16 * ct);
                __builtin_nontemporal_store(d,  dists + roff + 16 * ct);
            }
            // lane-local fold over the 4 column tiles
            float m4 = fmaxf(fmaxf(sim[0], sim[1]), fmaxf(sim[2], sim[3]));
            float s4 = __expf(sim[0] - m4) + __expf(sim[1] - m4) +
                       __expf(sim[2] - m4) + __expf(sim[3] - m4);
            // single 16-lane reduce: max, rescale once, sum
            float m16 = m4;
#pragma unroll
            for (int msk = 1; msk < 16; msk <<= 1)
                m16 = fmaxf(m16, __shfl_xor(m16, msk));
            float s16 = s4 * __expf(m4 - m16);
#pragma unroll
            for (int msk = 1; msk < 16; msk <<= 1)
                s16 += __shfl_xor(s16, msk);
            rm[rt][e] = m16;
            rs[rt][e] = s16;
        }
    }

    // combine across the two column-waves (wn=0 then wn=1), one partial/row
    if (wn == 0 && lm == 0) {
#pragma unroll
        for (int rt = 0; rt < 2; ++rt)
#pragma unroll
            for (int e = 0; e < 8; ++e) {
                int rl = 32 * wm + 16 * rt + e + 8 * lh;
                pm[rl] = rm[rt][e];
                ps[rl] = rs[rt][e];
            }
    }
    __syncthreads();
    if (wn == 1 && lm == 0) {
        int nb = gridDim.x;   // column blocks per row
#pragma unroll
        for (int rt = 0; rt < 2; ++rt)
#pragma unroll
            for (int e = 0; e < 8; ++e) {
                int rl = 32 * wm + 16 * rt + e + 8 * lh;
                float m = rm[rt][e], s = rs[rt][e];
                lse_merge(m, s, pm[rl], ps[rl]);
                size_t w = (size_t)(m_base + rl) * nb + blockIdx.x;
                wsm[w] = m;
                wss[w] = s;
            }
    }
}

// ---------------------------------------------------------------------------
// Kernel 3: merge per-row partials -> loss_i = logsumexp(sims[i,:]) - sims[i,0]
// (positive_indices cancel:  lse(s - c) - (s0 - c) = lse(s) - s0)
// ---------------------------------------------------------------------------
__global__ __launch_bounds__(256)
void uhg_row_combine_kernel(const float* __restrict__ wsm,
                            const float* __restrict__ wss,
                            const float* __restrict__ sims,
                            float* __restrict__ rowloss, int N, int nb) {
    int row = blockIdx.x * blockDim.x + threadIdx.x;
    if (row >= N) return;
    float m = -INFINITY, s = 0.f;
    const float* pmr = wsm + (size_t)row * nb;
    const float* psr = wss + (size_t)row * nb;
    for (int j = 0; j < nb; ++j) lse_merge(m, s, pmr[j], psr[j]);
    rowloss[row] = m + logf(s) - sims[(size_t)row * N];
}

// ---------------------------------------------------------------------------
// Kernel 4: final mean over rows -> d_out[0]
// ---------------------------------------------------------------------------
__global__ __launch_bounds__(256)
void uhg_final_reduce_kernel(const float* __restrict__ rowloss,
                             float* __restrict__ out, int N) {
    __shared__ float red[256];
    const int t = threadIdx.x;
    float sum = 0.f;
    for (int j = t; j < N; j += 256) sum += rowloss[j];
    red[t] = sum;
    __syncthreads();
    for (int o = 128; o > 0; o >>= 1) {
        if (t < o) red[t] += red[t + o];
        __syncthreads();
    }
    if (t == 0) out[0] = red[0] / (float)N;
}

// ---------------------------------------------------------------------------
extern "C" void kernel_launch(void* const* d_in, const int* in_sizes, int n_in,
                              void* d_out, int out_size, void* d_ws, size_t ws_size,
                              hipStream_t stream) {
    const float* q = (const float*)d_in[0];
    const float* k = (const float*)d_in[1];
    // d_in[2] (positive_indices) cancels analytically in the loss — unused.

    const int N = in_sizes[2];          // 8192
    const int D = in_sizes[0] / N;      // 64
    const int nb = N / TILE_N;          // 64 column blocks

    float* out   = (float*)d_out;
    float* sims  = out + 1;                       // [N,N]
    float* dists = out + 1 + (size_t)N * N;       // [N,N]

    float* qq      = (float*)d_ws;                // [N]
    float* kk      = qq + N;                      // [N]
    float* rowloss = kk + N;                      // [N]
    float* wsm     = rowloss + N;                 // [N*nb] partial maxes
    float* wss     = wsm + (size_t)N * nb;        // [N*nb] partial sums

    uhg_norms_kernel<<<(2 * N + 255) / 256, 256, 0, stream>>>(q, k, qq, kk, N, D);

    dim3 grid(N / TILE_N, N / TILE_M);            // 64 x 64 blocks
    uhg_gemm_kernel<<<grid, 256, 0, stream>>>(q, k, qq, kk, sims, dists,
                                              wsm, wss, N, D);

    uhg_row_combine_kernel<<<(N + 255) / 256, 256, 0, stream>>>(wsm, wss, sims,
                                                                rowloss, N, nb);
    uhg_final_reduce_kernel<<<1, 256, 0, stream>>>(rowloss, out, N);
}